// RELAX_All_45655502357108
// MI455X (gfx1250) — compile-verified
//
#include <hip/hip_runtime.h>
#include <math.h>

// Problem constants (from the reference)
#define HW     50176      // 224*224 pixels
#define NSAMP  1024       // masked samples
#define EPSW   1e-10f     // initial sum_of_weights

typedef __attribute__((ext_vector_type(2))) float v2f;
typedef __attribute__((ext_vector_type(8))) float v8f;

// ---------------------------------------------------------------------------
// Kernel 1: WMMA-based masked weighted sums (closed form of the online mean).
// One wave owns 32 consecutive pixels (two 16-px WMMA tiles sharing B):
//   A0/A1 (16x4 f32) = masks_a tiles  -> lane L: pixel = L&15 (+16), K = (L>>4)*2+v
//   BA    (4x16 f32) = col0 = sims[k], col1 = 1, rest 0
//   M0/M1 (16x4 f32) = masks_b tiles
//   BB    (4x16 f32) = col0 = sims_t, col1 = sims_c, col2 = sims_s, col3 = 1
// D += A*B accumulates per-pixel {Σ m·s, Σ m} via v_wmma_f32_16x16x4_f32.
// ---------------------------------------------------------------------------
__global__ __launch_bounds__(256) void relax_accum_kernel(
    const float* __restrict__ masks_a, const float* __restrict__ sims,
    const float* __restrict__ masks_b, const float* __restrict__ sims_t,
    const float* __restrict__ sims_c,  const float* __restrict__ sims_s,
    float* __restrict__ out)
{
    __shared__ float sh_s[NSAMP];          // sims
    __shared__ float sh_t[NSAMP];          // sims_texture
    __shared__ float sh_c[NSAMP];          // sims_color
    __shared__ float sh_h[NSAMP];          // sims_shape
    __shared__ float stageA[8][2][32];     // per-wave column->pixel transpose
    __shared__ float stageB[8][4][32];

    const int tid = threadIdx.x;

    // Cache per-sample weights in LDS once per block (uniform, 4 iterations)
    for (int i = tid; i < NSAMP; i += 256) {
        sh_s[i] = sims[i];
        sh_t[i] = sims_t[i];
        sh_c[i] = sims_c[i];
        sh_h[i] = sims_s[i];
    }
    __syncthreads();

    const int wave = tid >> 5;
    const int lane = tid & 31;
    const int n    = lane & 15;     // B/D column
    const int hi   = lane >> 4;     // lane half: K pair {0,1} or {2,3}
    const int k0   = 2 * hi;
    const int base = (blockIdx.x * 8 + wave) * 32;   // 32 pixels per wave
    const int p0   = base + n;

    // Hoisted B-operand selection state (loop body: 1 b64 LDS load + cndmask)
    const bool  selA = (n == 0);
    const bool  selB = (n < 3);
    const float cA   = (n == 1) ? 1.0f : 0.0f;
    const float cB   = (n == 3) ? 1.0f : 0.0f;
    const float* bsel = (n == 1) ? sh_c : (n == 2) ? sh_h : sh_t;

    const float* pa = masks_a + k0 * HW + p0;
    const float* pb = masks_b + k0 * HW + p0;

    v8f accA0 = {}, accA1 = {};   // masks_a tiles: col0 = Σ ma·s, col1 = Σ ma
    v8f accB0 = {}, accB1 = {};   // masks_b tiles: cols {Σ·t, Σ·c, Σ·sh, Σ mb}

    for (int i0 = 0; i0 < NSAMP; i0 += 4) {
        v2f A0, A1, M0, M1, BA, BB;
        // Streamed exactly once (411 MB > 192 MB L2) -> non-temporal.
        // 8 b32 loads off 2 base pointers via immediate offsets (fit IOFFSET).
        A0.x = __builtin_nontemporal_load(pa + 0);
        A1.x = __builtin_nontemporal_load(pa + 16);
        A0.y = __builtin_nontemporal_load(pa + HW);
        A1.y = __builtin_nontemporal_load(pa + HW + 16);
        M0.x = __builtin_nontemporal_load(pb + 0);
        M1.x = __builtin_nontemporal_load(pb + 16);
        M0.y = __builtin_nontemporal_load(pb + HW);
        M1.y = __builtin_nontemporal_load(pb + HW + 16);

        // Weights for samples (i0+k0, i0+k0+1): consecutive, 8B aligned.
        const int kk = i0 + k0;
        const float2 wa = *(const float2*)&sh_s[kk];
        const float2 wb = *(const float2*)&bsel[kk];
        BA.x = selA ? wa.x : cA;
        BA.y = selA ? wa.y : cA;
        BB.x = selB ? wb.x : cB;
        BB.y = selB ? wb.y : cB;

        accA0 = __builtin_amdgcn_wmma_f32_16x16x4_f32(
            false, A0, false, BA, (short)0, accA0, false, false);
        accA1 = __builtin_amdgcn_wmma_f32_16x16x4_f32(
            false, A1, false, BA, (short)0, accA1, false, false);
        accB0 = __builtin_amdgcn_wmma_f32_16x16x4_f32(
            false, M0, false, BB, (short)0, accB0, false, false);
        accB1 = __builtin_amdgcn_wmma_f32_16x16x4_f32(
            false, M1, false, BB, (short)0, accB1, false, false);

        pa += 4 * HW;
        pb += 4 * HW;
    }

    // D layout: lane holds column n, rows m = r + 8*hi. Transpose via LDS.
#pragma unroll
    for (int r = 0; r < 8; ++r) {
        const int m = r + 8 * hi;
        if (n < 2) {
            stageA[wave][n][m]      = accA0[r];
            stageA[wave][n][m + 16] = accA1[r];
        }
        if (n < 4) {
            stageB[wave][n][m]      = accB0[r];
            stageB[wave][n][m + 16] = accB1[r];
        }
    }
    __syncthreads();

    {
        const int m = lane;                               // pixel in 32-px tile
        const float denA = EPSW + stageA[wave][1][m];     // EPS + Σ ma
        const float denB = EPSW + stageB[wave][3][m];     // EPS + Σ mb
        out[0 * HW + base + m] = stageA[wave][0][m] / denA;  // imp
        out[1 * HW + base + m] = stageB[wave][0][m] / denB;  // imp_texture
        out[2 * HW + base + m] = stageB[wave][1][m] / denB;  // imp_color
        out[3 * HW + base + m] = stageB[wave][2][m] / denB;  // imp_shape
    }
}

// ---------------------------------------------------------------------------
// Kernel 2: deterministic per-plane mean and 1/std (ddof=1). One block/plane.
// ---------------------------------------------------------------------------
__global__ __launch_bounds__(256) void relax_stats_kernel(
    const float* __restrict__ out, float* __restrict__ stats)
{
    const int plane = blockIdx.x;
    const float* x = out + plane * HW;

    float s = 0.0f, q = 0.0f;
    for (int i = threadIdx.x; i < HW; i += 256) {
        float v = x[i];
        s += v;
        q += v * v;
    }

    __shared__ float ss[256];
    __shared__ float qs[256];
    ss[threadIdx.x] = s;
    qs[threadIdx.x] = q;
    __syncthreads();
    for (int ofs = 128; ofs > 0; ofs >>= 1) {
        if (threadIdx.x < ofs) {
            ss[threadIdx.x] += ss[threadIdx.x + ofs];
            qs[threadIdx.x] += qs[threadIdx.x + ofs];
        }
        __syncthreads();
    }
    if (threadIdx.x == 0) {
        const float nF   = (float)HW;
        const float mean = ss[0] / nF;
        const float var  = (qs[0] - ss[0] * ss[0] / nF) / (nF - 1.0f);
        stats[2 * plane + 0] = mean;
        stats[2 * plane + 1] = 1.0f / sqrtf(var);
    }
}

// ---------------------------------------------------------------------------
// Kernel 3: in-place normalization of d_out.
// ---------------------------------------------------------------------------
__global__ __launch_bounds__(256) void relax_norm_kernel(
    float* __restrict__ out, const float* __restrict__ stats)
{
    const int i = blockIdx.x * 256 + threadIdx.x;
    if (i < 4 * HW) {
        const int plane = i / HW;
        out[i] = (out[i] - stats[2 * plane]) * stats[2 * plane + 1];
    }
}

extern "C" void kernel_launch(void* const* d_in, const int* in_sizes, int n_in,
                              void* d_out, int out_size, void* d_ws, size_t ws_size,
                              hipStream_t stream) {
    (void)in_sizes; (void)n_in; (void)out_size; (void)ws_size;
    const float* masks_a = (const float*)d_in[0];
    const float* sims    = (const float*)d_in[1];
    const float* masks_b = (const float*)d_in[2];
    const float* sims_t  = (const float*)d_in[3];
    const float* sims_c  = (const float*)d_in[4];
    const float* sims_s  = (const float*)d_in[5];
    float* out   = (float*)d_out;
    float* stats = (float*)d_ws;   // 8 floats: {mean, rstd} x 4 planes

    // 50176 pixels / (8 waves * 32 pixels) = 196 blocks, exact tiling
    relax_accum_kernel<<<196, 256, 0, stream>>>(
        masks_a, sims, masks_b, sims_t, sims_c, sims_s, out);
    relax_stats_kernel<<<4, 256, 0, stream>>>(out, stats);
    relax_norm_kernel<<<(4 * HW + 255) / 256, 256, 0, stream>>>(out, stats);
}